// CorrelationFilter_39213051412494
// MI455X (gfx1250) — compile-verified
//
#include <hip/hip_runtime.h>

typedef float v2f  __attribute__((ext_vector_type(2)));
typedef float v8f  __attribute__((ext_vector_type(8)));
typedef unsigned int u32x4 __attribute__((ext_vector_type(4)));
typedef int   i32x4 __attribute__((ext_vector_type(4)));
typedef int   i32x8 __attribute__((ext_vector_type(8)));

#define NB    16          // batch
#define HSB   64          // source H/W
#define HT    16          // template H/W
#define CCH   256         // channels
#define HOUT  49          // output H/W
#define ROWF  (HSB*CCH)   // 16384 floats per source row (64 KB)
#define KTOT  (HT*CCH)    // 4096 = K of per-row GEMM
#define LDSF  20224       // padded LDS floats: (63*256 + 4095) + 1

// =====================================================================
// Kernel 1: per (b, source-row r) GEMM  partial[x, dy] = A(64x4096) * B(4096x16)
//   A[m,k] = srcrow[m*256 + k]  (sliding window, staged in LDS via TDM)
//   B[k,n] = tpl[b, dy=n, k]    (read from global / L2)
// 8 waves split K (512 each), 4 M-tiles per wave, v_wmma_f32_16x16x4_f32.
// Cross-wave reduce through LDS; store partial[b][r][x(64)][dy(16)] to ws.
// =====================================================================
__global__ void __launch_bounds__(256)
corr_row_gemm(const float* __restrict__ src,
              const float* __restrict__ tpl,
              float* __restrict__ ws) {
  extern __shared__ float lds[];
  const int r    = blockIdx.x;
  const int b    = blockIdx.y;
  const int tid  = threadIdx.x;
  const int lane = tid & 31;
  const int w    = tid >> 5;          // wave id 0..7 (wave32)

  // ---------------- stage source row (64 KB) into LDS ----------------
#if __has_builtin(__builtin_amdgcn_tensor_load_to_lds)
  {
    const unsigned sliceF = ROWF / 8;                 // 2048 floats per wave
    unsigned long long ga = (unsigned long long)src
                          + (unsigned long long)(b * HSB + r) * (ROWF * 4ull)
                          + (unsigned long long)w * (sliceF * 4u);
    unsigned la = (unsigned)(unsigned long long)(void*)lds + (unsigned)w * (sliceF * 4u);
    // D# group0: count=1, lds_addr, global_addr(57b), type=2
    u32x4 g0 = { 1u,
                 la,
                 (unsigned)(ga & 0xffffffffull),
                 (unsigned)((ga >> 32) & 0x01ffffffull) | 0x80000000u };
    // D# group1: data_size=4B(code 2); tensor_dim0=tile_dim0=stride0=sliceF;
    //            tensor_dim1=tile_dim1=1 (1-row tile)
    i32x8 g1 = { (int)(2u << 16),          // workgroup_mask=0 | data_size=2
                 (int)(sliceF << 16),      // tensor_dim0[15:0] @ bits 48..63
                 (int)(1u << 16),          // tensor_dim1[15:0] @ bits 80..95
                 (int)(sliceF << 16),      // tile_dim0 @ bits 112..127
                 1,                        // tile_dim1=1
                 (int)sliceF,              // tensor_dim0_stride[31:0]
                 (int)(sliceF << 16),      // tensor_dim1_stride[15:0]
                 0 };
    i32x4 gz4 = { 0, 0, 0, 0 };
#if defined(__clang_major__) && (__clang_major__ >= 23)
    i32x8 gz8 = { 0, 0, 0, 0, 0, 0, 0, 0 };
    __builtin_amdgcn_tensor_load_to_lds(g0, g1, gz4, gz4, gz8, 0);
#else
    __builtin_amdgcn_tensor_load_to_lds(g0, g1, gz4, gz4, 0);
#endif
    __builtin_amdgcn_s_wait_tensorcnt(0);
  }
#else
  { // cooperative fallback: 256 threads x 16 float4
    const float4* s4 = (const float4*)(src + (size_t)(b * HSB + r) * ROWF);
    float4* l4 = (float4*)lds;
#pragma unroll
    for (int j = 0; j < 16; ++j) l4[tid + j * 256] = s4[tid + j * 256];
  }
#endif
  __syncthreads();

  // ---------------- WMMA main loop ----------------
  const int n  = lane & 15;            // N index (dy) == A row-within-tile
  const int kl = (lane >> 4) << 1;     // lane-group K offset: 0 or 2
  const float* tplRow = tpl + (size_t)b * (HT * HT * CCH) + (size_t)n * KTOT;
  const float* aBase  = lds + n * CCH;

  v8f c0 = {}, c1 = {}, c2 = {}, c3 = {};
  const int k0 = w * (KTOT / 8);       // 512-wide K slice per wave

#pragma unroll 4
  for (int i = 0; i < (KTOT / 8) / 4; ++i) {      // 128 K-steps of 4
    const int k = k0 + i * 4 + kl;
    v2f bv = *(const v2f*)(tplRow + k);           // B: global (L2) b64
    v2f a0 = *(const v2f*)(aBase + k);            // A: LDS b64, 4 M-tiles
    v2f a1 = *(const v2f*)(aBase + 16 * CCH + k);
    v2f a2 = *(const v2f*)(aBase + 32 * CCH + k);
    v2f a3 = *(const v2f*)(aBase + 48 * CCH + k);
    c0 = __builtin_amdgcn_wmma_f32_16x16x4_f32(false, a0, false, bv, (short)0, c0, false, false);
    c1 = __builtin_amdgcn_wmma_f32_16x16x4_f32(false, a1, false, bv, (short)0, c1, false, false);
    c2 = __builtin_amdgcn_wmma_f32_16x16x4_f32(false, a2, false, bv, (short)0, c2, false, false);
    c3 = __builtin_amdgcn_wmma_f32_16x16x4_f32(false, a3, false, bv, (short)0, c3, false, false);
  }

  // ---------------- cross-wave reduction through LDS ----------------
  __syncthreads();                     // A-reads done; reuse lds as red buffer
#pragma unroll
  for (int v = 0; v < 8; ++v) {
    lds[(w * 4 + 0) * 256 + v * 32 + lane] = c0[v];
    lds[(w * 4 + 1) * 256 + v * 32 + lane] = c1[v];
    lds[(w * 4 + 2) * 256 + v * 32 + lane] = c2[v];
    lds[(w * 4 + 3) * 256 + v * 32 + lane] = c3[v];
  }
  __syncthreads();

  if (w < 4) {                         // wave t sums tile t across 8 waves
    const int t = w;
#pragma unroll
    for (int v = 0; v < 8; ++v) {
      float s = 0.f;
#pragma unroll
      for (int ww = 0; ww < 8; ++ww)
        s += lds[(ww * 4 + t) * 256 + v * 32 + lane];
      const int x = 16 * t + v + 8 * (lane >> 4);   // D layout: M = v + 8*(lane/16)
      ws[(((size_t)(b * HSB + r)) * HSB + x) * HT + n] = s;
    }
  }
}

// =====================================================================
// Kernel 2: out[b,y,x] = sum_c bias[y,x,b*256+c] + sum_dy ws[b][y+dy][x][dy]
// One wave per output element; shfl-xor wave reduction.
// =====================================================================
__global__ void __launch_bounds__(256)
corr_gather(const float* __restrict__ bias,
            const float* __restrict__ ws,
            float* __restrict__ out, int nOut) {
  const int lane = threadIdx.x & 31;
  const int w    = threadIdx.x >> 5;
  const int o    = blockIdx.x * 8 + w;
  if (o >= nOut) return;
  const int b  = o / (HOUT * HOUT);
  const int yx = o % (HOUT * HOUT);
  const int y  = yx / HOUT;
  const int x  = yx % HOUT;

  float s = 0.f;
  const float* bp = bias + (size_t)yx * (NB * CCH) + b * CCH;
#pragma unroll
  for (int j = 0; j < 8; ++j) s += bp[lane + 32 * j];

  if (lane < HT)
    s += ws[(((size_t)(b * HSB + y + lane)) * HSB + x) * HT + lane];

#pragma unroll
  for (int off = 16; off; off >>= 1) s += __shfl_xor(s, off, 32);
  if (lane == 0) out[o] = s;
}

// =====================================================================
extern "C" void kernel_launch(void* const* d_in, const int* in_sizes, int n_in,
                              void* d_out, int out_size, void* d_ws, size_t ws_size,
                              hipStream_t stream) {
  const float* src  = (const float*)d_in[0];   // (16,64,64,256) f32
  const float* tpl  = (const float*)d_in[1];   // (16,16,16,256) f32
  const float* bias = (const float*)d_in[2];   // (1,49,49,4096) f32
  float* out = (float*)d_out;                  // (16,49,49,1)   f32
  float* ws  = (float*)d_ws;                   // needs 16*64*64*16*4 = 4 MB

  dim3 grid1(HSB, NB);                         // one WG per (r, b)
  corr_row_gemm<<<grid1, 256, LDSF * sizeof(float), stream>>>(src, tpl, ws);

  const int nOut = NB * HOUT * HOUT;           // 38416
  const int blocks = (nOut + 7) / 8;
  corr_gather<<<blocks, 256, 0, stream>>>(bias, ws, out, nOut);
}